// TPSPathModule_23201413333098
// MI455X (gfx1250) — compile-verified
//
#include <hip/hip_runtime.h>
#include <hip/hip_bf16.h>

typedef __attribute__((ext_vector_type(2))) float v2f;
typedef __attribute__((ext_vector_type(8))) float v8f;

#define NCTRL 256
#define KFEAT 4096
#define NEVAL 8192
#define NSEG  255          // NCTRL-1
#define MINT  254          // interior knots

// ---------------------------------------------------------------------------
// Kernel 1: natural cubic spline fit — Thomas solve of the tridiagonal
// system R g = Q^T y for all 4096 feature columns in parallel.
// Uniform knots => elimination coefficients are column-independent: compute
// once into LDS. gamma written to workspace, rows 0 and 255 are zero (natural
// boundary conditions).
// ---------------------------------------------------------------------------
__global__ __launch_bounds__(256) void spline_fit_kernel(
    const float* __restrict__ V,   // (256, 4096)
    float* __restrict__ G)         // (256, 4096) gamma out (workspace)
{
    __shared__ float s_cp[MINT];
    __shared__ float s_inv[MINT];

    const float h    = 1.0f / 255.0f;
    const float off  = h / 6.0f;
    const float diag = 2.0f * h / 3.0f;

    if (threadIdx.x == 0) {
        float cp_prev = 0.0f;
        for (int j = 0; j < MINT; ++j) {
            float denom = diag - off * cp_prev;
            float inv   = 1.0f / denom;
            float cp    = off * inv;
            s_inv[j] = inv;
            s_cp[j]  = cp;
            cp_prev  = cp;
        }
    }
    __syncthreads();

    const int k = blockIdx.x * blockDim.x + threadIdx.x;   // feature column
    if (k >= KFEAT) return;

    // forward elimination: d'_j stored at gamma row j+1
    float vprev = V[0 * KFEAT + k];
    float vcur  = V[1 * KFEAT + k];
    float dp = 0.0f;
    for (int j = 0; j < MINT; ++j) {
        float vnext = V[(j + 2) * KFEAT + k];
        float dj = 255.0f * ((vnext - vcur) - (vcur - vprev));  // Q^T y
        dp = (dj - off * dp) * s_inv[j];
        G[(j + 1) * KFEAT + k] = dp;
        vprev = vcur;
        vcur  = vnext;
    }
    // natural BCs
    G[0 * KFEAT + k]            = 0.0f;
    G[(NCTRL - 1) * KFEAT + k]  = 0.0f;

    // back substitution: g_j = d'_j - cp_j * g_{j+1}
    float gnext = G[MINT * KFEAT + k];   // g_{253} = d'_{253} already stored
    for (int j = MINT - 2; j >= 0; --j) {
        float dpj = G[(j + 1) * KFEAT + k];
        float gj  = dpj - s_cp[j] * gnext;
        G[(j + 1) * KFEAT + k] = gj;
        gnext = gj;
    }
}

// ---------------------------------------------------------------------------
// Kernel 2: spline evaluation as WMMA. One wave owns a 16-eval-row x 128-
// feature strip. 16 consecutive eval points touch at most 3 knots
// {j0, j0+1, j0+2}, so out_tile = A0(16x4) x B0(4x16) + A1(16x4) x B1(4x16)
// via two chained V_WMMA_F32_16X16X4_F32. A matrices reused across the 8
// feature tiles; B tiles are 3 rows of y / gamma (L2-resident, 8 MB total).
// Output (134 MB) streamed with nontemporal stores.
// ---------------------------------------------------------------------------
__global__ __launch_bounds__(128) void spline_eval_wmma_kernel(
    const float* __restrict__ V,   // y_hat = values (lam = 0), (256,4096)
    const float* __restrict__ G,   // gamma, (256,4096)
    float* __restrict__ out)       // (8192,4096)
{
    const int lane    = threadIdx.x & 31;
    const int wave    = threadIdx.x >> 5;
    const int halfsel = lane >> 4;        // 0: lanes 0-15, 1: lanes 16-31
    const int lrow    = lane & 15;

    const int et    = blockIdx.x;                 // eval tile, 0..511
    const int m0    = et * 16;                    // first eval row of tile
    const int nbase = blockIdx.y * 512 + wave * 128;

    // ---- per-lane basis coefficients (A-matrix lane layout: M = lane%16,
    //      lane supplies K = 2*(lane/16) in v0 and K+1 in v1) ----
    const int ie  = m0 + lrow;                       // this lane's eval row
    int idx = (ie * 255) / 8191; if (idx > NSEG - 1) idx = NSEG - 1;   // exact searchsorted
    int j0  = (m0 * 255) / 8191; if (j0  > NSEG - 1) j0  = NSEG - 1;   // tile's first segment
    const int di = idx - j0;                         // 0 or 1 within tile

    const float r  = (float)(ie * 255) / 8191.0f;    // 255*t, numerator exact in f32
    const float a  = (float)(idx + 1) - r;           // (x_hi - t)/h
    const float b  = 1.0f - a;
    const float w  = 1.0f / (255.0f * 255.0f * 6.0f);// h^2/6
    const float ca = (a * a * a - a) * w;
    const float cb = (b * b * b - b) * w;

    const int kk = halfsel * 2;                      // K index of v0 slot
    auto coef = [&](int K, float pa, float pb) -> float {
        return (K == di) ? pa : ((K == di + 1) ? pb : 0.0f);
    };
    v2f A0, A1;
    A0.x = coef(kk,     a,  b);  A0.y = coef(kk + 1,  a,  b);
    A1.x = coef(kk,     ca, cb); A1.y = coef(kk + 1,  ca, cb);

    // ---- B-matrix knot rows for this lane (K row = 2*(lane/16) in v0, +1 in v1);
    //      clamp: K=3 column (and K=2 at the last segment) has zero A coeffs ----
    int rk0 = j0 + kk;     if (rk0 > NCTRL - 1) rk0 = NCTRL - 1;
    int rk1 = j0 + kk + 1; if (rk1 > NCTRL - 1) rk1 = NCTRL - 1;

    const int rowbase = m0 + halfsel * 8;            // C/D layout: VGPR r -> M=r / r+8

    for (int ft = 0; ft < 8; ++ft) {
        const int col = nbase + ft * 16 + lrow;

        v2f B0, B1;
        B0.x = V[rk0 * KFEAT + col];
        B0.y = V[rk1 * KFEAT + col];
        B1.x = G[rk0 * KFEAT + col];
        B1.y = G[rk1 * KFEAT + col];

        v8f c = {};
        // 8 args: (neg_a, A, neg_b, B, c_mod, C, reuse_a, reuse_b)
        c = __builtin_amdgcn_wmma_f32_16x16x4_f32(false, A0, false, B0,
                                                  (short)0, c, false, false);
        c = __builtin_amdgcn_wmma_f32_16x16x4_f32(false, A1, false, B1,
                                                  (short)0, c, false, false);

        #pragma unroll
        for (int rr = 0; rr < 8; ++rr) {
            __builtin_nontemporal_store(
                c[rr], &out[(size_t)(rowbase + rr) * KFEAT + col]);
        }
    }
}

// ---------------------------------------------------------------------------
extern "C" void kernel_launch(void* const* d_in, const int* in_sizes, int n_in,
                              void* d_out, int out_size, void* d_ws, size_t ws_size,
                              hipStream_t stream) {
    const float* values = (const float*)d_in[0];   // (256, 4096) f32
    float* out   = (float*)d_out;                  // (8192, 4096) f32
    float* gamma = (float*)d_ws;                   // 256*4096*4 = 4 MB scratch

    // Phase 1: tridiagonal fit (one thread per feature column)
    spline_fit_kernel<<<KFEAT / 256, 256, 0, stream>>>(values, gamma);

    // Phase 2: WMMA evaluation. 512 eval tiles x 8 feature groups,
    // 4 waves/block, each wave -> 16 rows x 128 features.
    dim3 grid(NEVAL / 16, KFEAT / 512);
    spline_eval_wmma_kernel<<<grid, 128, 0, stream>>>(values, gamma, out);
}